// TCNBlock_33483565039654
// MI455X (gfx1250) — compile-verified
//
#include <hip/hip_runtime.h>
#include <cstdint>

#define B_   4
#define C_   128
#define H_   512
#define T_   8000
#define DIL_ 8

typedef float v2f __attribute__((ext_vector_type(2)));
typedef float v4f __attribute__((ext_vector_type(4)));
typedef float v8f __attribute__((ext_vector_type(8)));

__device__ __forceinline__ v8f wmma4(v2f a, v2f b, v8f c) {
  // V_WMMA_F32_16X16X4_F32 : exact fp32 16x16 tile, K=4 per issue
  return __builtin_amdgcn_wmma_f32_16x16x4_f32(false, a, false, b, (short)0, c, false, false);
}

// ---------------------------------------------------------------------------
// Fused WMMA GEMM:  Y[b,m,t] = op( A[m,:] . Bm[b,:,t] + bias[m] )
//   NORM_B      : apply cumLN affine to B operand on load (gemm3)
//   PRELU_SUMS  : PReLU + atomic column sums for next cumLN (gemm1/2)
//   ADD_X       : residual add (gemm3)
// Block = 128 threads (4 waves), 2(M)x2(N) waves, block tile 128M x 64N,
// wave tile 64M x 32N: per K-step 8 WMMA vs 4xb64 + 4xb32 loads (1.0 vmem/wmma)
// ---------------------------------------------------------------------------
template<bool NORM_B, bool PRELU_SUMS, bool ADD_X>
__global__ __launch_bounds__(128)
void k_gemm(const float* __restrict__ A,      // Mrows x K row-major
            const float* __restrict__ Bm,     // (B, K, T_)
            const float* __restrict__ bias,   // Mrows
            const float* __restrict__ prelu_a,// scalar
            const float* __restrict__ nw,     // K (norm gain)
            const float* __restrict__ nb,     // K (norm bias)
            const float* __restrict__ mean,   // (B, T_)
            const float* __restrict__ istd,   // (B, T_)
            const float* __restrict__ Xres,   // (B, Mrows, T_)
            float* __restrict__ Y,            // (B, Mrows, T_)
            float* __restrict__ s1, float* __restrict__ s2, // (B, T_)
            int Mrows, int K)
{
  int blk = blockIdx.x;
  const int NT = T_ / 64;               // 125
  int nblk = blk % NT; blk /= NT;
  int Mtiles = Mrows >> 7;              // 128 M rows per block
  int mblk = blk % Mtiles; blk /= Mtiles;
  int b = blk;

  int tid  = threadIdx.x;
  int lane = tid & 31;
  int wave = tid >> 5;                  // 0..3 : (wave>>1)=M half, (wave&1)=N half
  int lm   = lane & 15;                 // row/col within 16
  int hk   = lane >> 4;                 // K-half select for A/B frags

  int m0  = mblk * 128 + (wave >> 1) * 64;
  int n0  = nblk * 64 + (wave & 1) * 32;
  int tc0 = n0 + lm;
  int tc1 = tc0 + 16;

  const float* Bb   = Bm + (size_t)b * K * T_;
  const float* Abase = A + (size_t)(m0 + lm) * K + 2 * hk;   // frag i: + i*16*K

  float mt0 = 0.f, it0 = 0.f, mt1 = 0.f, it1 = 0.f;
  if (NORM_B) {   // per-column stats: hoisted out of K loop
    mt0 = mean[b * T_ + tc0]; it0 = istd[b * T_ + tc0];
    mt1 = mean[b * T_ + tc1]; it1 = istd[b * T_ + tc1];
  }

  v8f acc[4][2] = {};

  for (int k = 0; k < K; k += 4) {
    v2f a[4];
    #pragma unroll
    for (int i = 0; i < 4; ++i)
      a[i] = *(const v2f*)(Abase + (size_t)i * 16 * K + k);

    int kk = k + 2 * hk;
    const float* brow0 = Bb + (size_t)kk * T_;
    const float* brow1 = brow0 + T_;
    float b0x = brow0[tc0], b0y = brow1[tc0];
    float b1x = brow0[tc1], b1y = brow1[tc1];
    if (NORM_B) {
      float w0 = nw[kk], w1 = nw[kk + 1];
      float c0 = nb[kk], c1 = nb[kk + 1];
      b0x = w0 * (b0x - mt0) * it0 + c0;
      b0y = w1 * (b0y - mt0) * it0 + c1;
      b1x = w0 * (b1x - mt1) * it1 + c0;
      b1y = w1 * (b1y - mt1) * it1 + c1;
    }
    v2f bf0 = {b0x, b0y};
    v2f bf1 = {b1x, b1y};
    #pragma unroll
    for (int i = 0; i < 4; ++i) {
      acc[i][0] = wmma4(a[i], bf0, acc[i][0]);
      acc[i][1] = wmma4(a[i], bf1, acc[i][1]);
    }
  }

  float aS = 0.f;
  if (PRELU_SUMS) aS = prelu_a[0];

  #pragma unroll
  for (int jn = 0; jn < 2; ++jn) {
    int t = jn ? tc1 : tc0;
    float ps1 = 0.f, ps2 = 0.f;
    #pragma unroll
    for (int im = 0; im < 4; ++im) {
      int Mb = m0 + im * 16 + 8 * hk;        // D layout: lane half owns rows r / r+8
      v4f blo = *(const v4f*)(bias + Mb);
      v4f bhi = *(const v4f*)(bias + Mb + 4);
      v8f v = acc[im][jn];
      #pragma unroll
      for (int r = 0; r < 8; ++r) {
        float y = v[r] + (r < 4 ? blo[r & 3] : bhi[r & 3]);
        if (PRELU_SUMS) {
          y = (y >= 0.f) ? y : aS * y;
          ps1 += y; ps2 += y * y;
        }
        if (ADD_X) y += Xres[((size_t)b * Mrows + Mb + r) * T_ + t];
        Y[((size_t)b * Mrows + Mb + r) * T_ + t] = y;
      }
    }
    if (PRELU_SUMS) {   // reduced over 4 M-subtiles in registers: 4x fewer atomics
      atomicAdd(&s1[b * T_ + t], ps1);
      atomicAdd(&s2[b * T_ + t], ps2);
    }
  }
}

// ---------------------------------------------------------------------------
// Cumulative scan over T of per-timestep channel sums -> mean / rstd per (b,t)
// One block per batch; wave-level shfl_up scan + cross-wave LDS + carry.
// ---------------------------------------------------------------------------
__global__ __launch_bounds__(256)
void k_scan(const float* __restrict__ s1, const float* __restrict__ s2,
            float* __restrict__ mean, float* __restrict__ istd)
{
  int b = blockIdx.x;
  int tid = threadIdx.x, lane = tid & 31, wid = tid >> 5;
  __shared__ float ws1[8], ws2[8];
  __shared__ float carry1, carry2;
  if (tid == 0) { carry1 = 0.f; carry2 = 0.f; }
  __syncthreads();

  for (int t0 = 0; t0 < T_; t0 += 256) {
    int t = t0 + tid;
    float v1 = (t < T_) ? s1[b * T_ + t] : 0.f;
    float v2 = (t < T_) ? s2[b * T_ + t] : 0.f;
    #pragma unroll
    for (int d = 1; d < 32; d <<= 1) {
      float u1 = __shfl_up(v1, d, 32);
      float u2 = __shfl_up(v2, d, 32);
      if (lane >= d) { v1 += u1; v2 += u2; }
    }
    if (lane == 31) { ws1[wid] = v1; ws2[wid] = v2; }
    __syncthreads();
    float o1 = carry1, o2 = carry2;
    for (int w = 0; w < wid; ++w) { o1 += ws1[w]; o2 += ws2[w]; }
    v1 += o1; v2 += o2;
    if (t < T_) {
      float cnt = (float)(t + 1) * (float)H_;
      float m   = v1 / cnt;
      float var = v2 / cnt - m * m;
      mean[b * T_ + t] = m;
      istd[b * T_ + t] = rsqrtf(var + 1e-6f);
    }
    __syncthreads();
    if (tid == 0) {
      float c1 = carry1, c2 = carry2;
      for (int w = 0; w < 8; ++w) { c1 += ws1[w]; c2 += ws2[w]; }
      carry1 = c1; carry2 = c2;
    }
    __syncthreads();
  }
}

// ---------------------------------------------------------------------------
// Fused: cumLN-normalize (gln1) + causal dilated depthwise conv (k=3, d=8)
// zero-padded taps at t-16, t-8, t (pad applied after norm, so tap<0 -> 0)
// ---------------------------------------------------------------------------
__global__ __launch_bounds__(256)
void k_dwnorm(const float* __restrict__ Y1, const float* __restrict__ mean,
              const float* __restrict__ istd,
              const float* __restrict__ gw, const float* __restrict__ gb,
              const float* __restrict__ w,  const float* __restrict__ dwb,
              float* __restrict__ Z)
{
  size_t idx = (size_t)blockIdx.x * 256 + threadIdx.x;
  const size_t total = (size_t)B_ * H_ * T_;
  if (idx >= total) return;
  int t = (int)(idx % T_);
  size_t bh = idx / T_;
  int h = (int)(bh % H_);
  int b = (int)(bh / H_);

  float g = gw[h], gbv = gb[h];
  float accv = dwb[h];
  const float* yrow = Y1 + bh * T_;
  const float* mrow = mean + (size_t)b * T_;
  const float* irow = istd + (size_t)b * T_;
  #pragma unroll
  for (int j = 0; j < 3; ++j) {
    int tau = t - 2 * DIL_ + j * DIL_;
    if (tau >= 0) {
      float zn = g * (yrow[tau] - mrow[tau]) * irow[tau] + gbv;
      accv += w[h * 3 + j] * zn;
    }
  }
  Z[idx] = accv;
}

// ---------------------------------------------------------------------------
extern "C" void kernel_launch(void* const* d_in, const int* in_sizes, int n_in,
                              void* d_out, int out_size, void* d_ws, size_t ws_size,
                              hipStream_t stream) {
  (void)in_sizes; (void)n_in; (void)out_size; (void)ws_size;

  const float* x   = (const float*)d_in[0];
  const float* w1  = (const float*)d_in[1];
  const float* b1  = (const float*)d_in[2];
  const float* a1  = (const float*)d_in[3];
  const float* g1w = (const float*)d_in[4];
  const float* g1b = (const float*)d_in[5];
  const float* dww = (const float*)d_in[6];
  const float* dwb = (const float*)d_in[7];
  const float* pww = (const float*)d_in[8];
  const float* pwb = (const float*)d_in[9];
  const float* a2  = (const float*)d_in[10];
  const float* g2w = (const float*)d_in[11];
  const float* g2b = (const float*)d_in[12];
  const float* w2  = (const float*)d_in[13];
  const float* b2  = (const float*)d_in[14];
  float* out = (float*)d_out;

  float* ws = (float*)d_ws;
  const size_t NBIG = (size_t)B_ * H_ * T_;      // 16.384 M floats
  float* big0 = ws;                               // Y1, later reused for U
  float* big1 = big0 + NBIG;                      // Z
  float* s1   = big1 + NBIG;                      // (B,T) column sums
  float* s2   = s1 + (size_t)B_ * T_;
  float* mn   = s2 + (size_t)B_ * T_;
  float* rs   = mn + (size_t)B_ * T_;

  dim3 blk(128);
  const int gridHH = B_ * (H_ / 128) * (T_ / 64); // 2000
  const int gridC  = B_ * (C_ / 128) * (T_ / 64); // 500

  // ---- stage 1: Y1 = prelu(W1 @ X + b1), column sums ----
  hipMemsetAsync(s1, 0, 2 * (size_t)B_ * T_ * sizeof(float), stream);
  k_gemm<false, true, false><<<gridHH, blk, 0, stream>>>(
      w1, x, b1, a1, nullptr, nullptr, nullptr, nullptr, nullptr,
      big0, s1, s2, H_, C_);
  k_scan<<<B_, 256, 0, stream>>>(s1, s2, mn, rs);

  // ---- stage 2: Z = dwconv(cumLN1(Y1)) + dw_b ----
  k_dwnorm<<<(unsigned)(NBIG / 256), 256, 0, stream>>>(
      big0, mn, rs, g1w, g1b, dww, dwb, big1);

  // ---- stage 3: U = prelu(Wpw @ Z + bpw), column sums ----
  hipMemsetAsync(s1, 0, 2 * (size_t)B_ * T_ * sizeof(float), stream);
  k_gemm<false, true, false><<<gridHH, blk, 0, stream>>>(
      pww, big1, pwb, a2, nullptr, nullptr, nullptr, nullptr, nullptr,
      big0, s1, s2, H_, H_);
  k_scan<<<B_, 256, 0, stream>>>(s1, s2, mn, rs);

  // ---- stage 4: out = W2 @ cumLN2(U) + b2 + x  (norm fused into B load) ----
  k_gemm<true, false, true><<<gridC, blk, 0, stream>>>(
      w2, big0, b2, nullptr, g2w, g2b, mn, rs, x,
      out, nullptr, nullptr, C_, H_);
}